// MultiRoIPool3d_46084999086896
// MI455X (gfx1250) — compile-verified
//
#include <hip/hip_runtime.h>
#include <math.h>
#include <stdint.h>

// Problem constants (from reference setup_inputs)
#define B_    2
#define P_    8192
#define R_    128
#define C_    64
#define V3_   27
#define V5_   125
#define VT_   152          // 27 + 125, concat order: scale-3 voxels then scale-5
#define TILE_ 1024         // points staged per LDS tile
#define NTILES_ (P_ / TILE_)
#define BLOCK_ 256

// ---- CDNA5 async global->LDS path (guarded; falls back to sync copy) ----
typedef int v4i __attribute__((ext_vector_type(4)));
typedef __attribute__((address_space(1))) v4i* g4p_t;   // global int4*
typedef __attribute__((address_space(3))) v4i* l4p_t;   // LDS int4*

#if defined(__gfx1250__) && __has_builtin(__builtin_amdgcn_global_load_async_to_lds_b128)
#define HAVE_ASYNC_LDS 1
#endif

static __device__ __forceinline__ void wait_async_done() {
#if defined(HAVE_ASYNC_LDS)
#if __has_builtin(__builtin_amdgcn_s_wait_asynccnt)
  __builtin_amdgcn_s_wait_asynccnt(0);
#else
  asm volatile("s_wait_asynccnt 0" ::: "memory");
#endif
#endif
}

// Float atomic-max on an LDS cell that holds float bits in a u32.
// Sign-aware int trick: non-negative floats order as signed ints (ds_max_i32),
// negative floats order inversely as unsigned ints (ds_min_u32).
// Init value must be 0xFF800000 (-inf bits). Handles -0.0 correctly.
static __device__ __forceinline__ void ldsAtomicMaxF(unsigned int* addr, float val) {
  unsigned int u = __float_as_uint(val);
  if (u & 0x80000000u) {
    atomicMin(addr, u);                 // ds_min_u32
  } else {
    atomicMax((int*)addr, (int)u);      // ds_max_i32
  }
}

// clip(floor(num/den * fo), 0, fo-1)  -- matches jnp.clip(floor(...)) semantics
static __device__ __forceinline__ int vox1d(float num, float den, float fo) {
  float f = floorf(num / den * fo);
  f = fminf(fmaxf(f, 0.0f), fo - 1.0f);
  return (int)f;
}

__global__ __launch_bounds__(BLOCK_) void MultiRoIPool3d_kernel(
    const float* __restrict__ pts,    // (B, P, 3)
    const float* __restrict__ feats,  // (B, C, P)
    const float* __restrict__ rois,   // (B, R, 7)
    float* __restrict__ out)          // (B*R, C, 152)
{
  // Double-buffered point tiles (float4-packed, 12 KB each) + pooled tensor.
  __shared__ float4      ptile[2][TILE_ * 3 / 4];   // 2 * 12288 B
  __shared__ unsigned int pool[C_ * VT_];           // 38912 B, float bits

  const int rg = blockIdx.x;          // global ROI id: b*R + r
  const int b  = rg / R_;

  // ROI params (uniform across block -> scalar loads)
  const float* rp = rois + (size_t)rg * 7;
  const float cx = rp[0], cy = rp[1], cz = rp[2];
  const float dx = rp[3], dy = rp[4], dz = rp[5];
  const float rz = rp[6];
  const float cr = cosf(rz);
  const float sr = sinf(rz);
  const float hx = 0.5f * dx, hy = 0.5f * dy;

  // Init pooled buffer to -inf bits.
  for (int i = threadIdx.x; i < C_ * VT_; i += BLOCK_) pool[i] = 0xFF800000u;

  // Tile stager: copy TILE_ points (TILE_*3 floats, 16B-aligned range) into LDS.
  auto stage = [&](int t, int bufi) {
    const float4* gsrc = (const float4*)(pts + ((size_t)b * P_ + (size_t)t * TILE_) * 3);
    float4* ldst = ptile[bufi];
    for (int i = threadIdx.x; i < TILE_ * 3 / 4; i += BLOCK_) {
#if defined(HAVE_ASYNC_LDS)
      __builtin_amdgcn_global_load_async_to_lds_b128(
          (g4p_t)(gsrc + i), (l4p_t)(ldst + i), 0, 0);
#else
      ldst[i] = gsrc[i];
#endif
    }
  };

  stage(0, 0);  // prefetch first tile

  const float* fbase = feats + (size_t)b * C_ * P_;
  const int lane = threadIdx.x & 31;

  for (int t = 0; t < NTILES_; ++t) {
    const int cur = t & 1;
    wait_async_done();      // my wave's async copies into ptile[cur] are in LDS
    __syncthreads();        // everyone's copies visible; previous buffer free

    if (t + 1 < NTILES_) stage(t + 1, cur ^ 1);   // overlap next copy w/ compute

    const float* tp = (const float*)ptile[cur];
    // TILE_/BLOCK_ = 4 full iterations: every lane always active (no tail),
    // so wave-uniform ballot/shfl below is safe.
    for (int j = threadIdx.x; j < TILE_; j += BLOCK_) {
      const float px = tp[3 * j + 0];
      const float py = tp[3 * j + 1];
      const float pz = tp[3 * j + 2];
      const float sx = px - cx, sy = py - cy, lz = pz - cz;
      const float lx =  sx * cr + sy * sr;
      const float ly = -sx * sr + sy * cr;

      // in-box test is identical for both scales (z is bottom-centered)
      const bool in = (fabsf(lx) <= hx) && (fabsf(ly) <= hy) &&
                      (lz >= 0.0f) && (lz <= dz);

      // Compute voxel ids unconditionally (clamped => well-defined even when
      // !in) so shfl sources below are always live registers.
      const int vx3 = vox1d(lx + hx, dx, 3.0f);
      const int vy3 = vox1d(ly + hy, dy, 3.0f);
      const int vz3 = vox1d(lz,      dz, 3.0f);
      const int vx5 = vox1d(lx + hx, dx, 5.0f);
      const int vy5 = vox1d(ly + hy, dy, 5.0f);
      const int vz5 = vox1d(lz,      dz, 5.0f);
      const int v3 = (vx3 * 3 + vy3) * 3 + vz3;           // [0, 27)
      const int v5 = V3_ + (vx5 * 5 + vy5) * 5 + vz5;     // [27, 152)
      const int p  = t * TILE_ + j;

      // Wave-cooperative drain of sparse in-box points: 32 lanes cover the
      // 64 channels (2 each) of one point at a time -> full MLP, no serial
      // 64-deep per-lane loop, atomics hit distinct LDS addresses.
      unsigned long long mask = __ballot(in);
      while (mask) {
        const int src = __ffsll((long long)mask) - 1;
        mask &= mask - 1;
        const int pp  = __shfl(p,  src);
        const int vv3 = __shfl(v3, src);
        const int vv5 = __shfl(v5, src);
        const float f0 = fbase[(size_t)lane        * P_ + pp];
        const float f1 = fbase[(size_t)(lane + 32) * P_ + pp];
        ldsAtomicMaxF(&pool[lane        * VT_ + vv3], f0);
        ldsAtomicMaxF(&pool[lane        * VT_ + vv5], f0);
        ldsAtomicMaxF(&pool[(lane + 32) * VT_ + vv3], f1);
        ldsAtomicMaxF(&pool[(lane + 32) * VT_ + vv5], f1);
      }
    }
    __syncthreads();        // tile fully pooled before its buffer is re-staged
  }

  // Stream out: (rg, c, v) contiguous; empty voxels (-inf bits) -> 0.0
  float* op = out + (size_t)rg * (C_ * VT_);
  for (int i = threadIdx.x; i < C_ * VT_; i += BLOCK_) {
    const unsigned int u = pool[i];
    op[i] = (u == 0xFF800000u) ? 0.0f : __uint_as_float(u);
  }
}

extern "C" void kernel_launch(void* const* d_in, const int* in_sizes, int n_in,
                              void* d_out, int out_size, void* d_ws, size_t ws_size,
                              hipStream_t stream) {
  const float* pts   = (const float*)d_in[0];  // points_xyz (B,P,3)
  const float* feats = (const float*)d_in[1];  // features   (B,C,P)
  const float* rois  = (const float*)d_in[2];  // rois       (B,R,7)
  float* out = (float*)d_out;                  // (B*R, C, 152)
  (void)in_sizes; (void)n_in; (void)out_size; (void)d_ws; (void)ws_size;

  dim3 grid(B_ * R_);
  dim3 block(BLOCK_);
  hipLaunchKernelGGL(MultiRoIPool3d_kernel, grid, block, 0, stream,
                     pts, feats, rois, out);
}